// StateValue_84138409328803
// MI455X (gfx1250) — compile-verified
//
#include <hip/hip_runtime.h>
#include <cstdint>
#include <cstddef>

typedef __attribute__((ext_vector_type(2))) float v2f;
typedef __attribute__((ext_vector_type(8))) float v8f;

#define HIDDEN 1024
#define DEXP   4096
#define NEXP   16
#define BATCH  4096
#define LN_EPS 1e-5f

#define BT     16                       // batch rows per workgroup
#define WAVES  16                       // waves per block (wave32)
#define TPB    (WAVES * 32)             // 512 threads
#define TPW    ((DEXP / 16) / WAVES)    // 16 N-tiles (16 cols each) per wave

// ---------------------------------------------------------------------------
// Kernel 1: gating = softmax(x @ Wg^T + bg)   -> gate[B][16]
// One wave32 per batch row.
// ---------------------------------------------------------------------------
__global__ __launch_bounds__(256)
void moe_gating_kernel(const float* __restrict__ x,
                       const float* __restrict__ Wg,
                       const float* __restrict__ bg,
                       float* __restrict__ gate) {
    const int wave = threadIdx.x >> 5;
    const int lane = threadIdx.x & 31;
    const int b    = blockIdx.x * (blockDim.x >> 5) + wave;

    float acc[NEXP];
#pragma unroll
    for (int e = 0; e < NEXP; ++e) acc[e] = 0.0f;

    const float* xr = x + (size_t)b * HIDDEN;
    for (int h = lane; h < HIDDEN; h += 32) {
        float xv = xr[h];
#pragma unroll
        for (int e = 0; e < NEXP; ++e)
            acc[e] = fmaf(xv, Wg[e * HIDDEN + h], acc[e]);
    }
#pragma unroll
    for (int e = 0; e < NEXP; ++e) {
#pragma unroll
        for (int m = 16; m >= 1; m >>= 1)
            acc[e] += __shfl_xor(acc[e], m, 32);
    }
    if (lane == 0) {
        float mx = -3.4e38f;
#pragma unroll
        for (int e = 0; e < NEXP; ++e) { acc[e] += bg[e]; mx = fmaxf(mx, acc[e]); }
        float s = 0.0f;
#pragma unroll
        for (int e = 0; e < NEXP; ++e) { acc[e] = __expf(acc[e] - mx); s += acc[e]; }
        float inv = 1.0f / s;
#pragma unroll
        for (int e = 0; e < NEXP; ++e)
            gate[(size_t)b * NEXP + e] = acc[e] * inv;
    }
}

// ---------------------------------------------------------------------------
// Kernel 2: fused expert GEMM + bias + layernorm + relu + W2-dot + tanh*gate.
// Block = (16 batch rows, 1 expert). 16 waves; wave w owns D columns
// [w*256, w*256+256) as 16 WMMA accumulator tiles held in VGPRs the whole time.
// ---------------------------------------------------------------------------
__global__ __launch_bounds__(TPB, 1)
void moe_expert_kernel(const float* __restrict__ x,
                       const float* __restrict__ W1,
                       const float* __restrict__ b1,
                       const float* __restrict__ ln_g,
                       const float* __restrict__ ln_b,
                       const float* __restrict__ W2,
                       const float* __restrict__ b2,
                       const float* __restrict__ gate,
                       float* __restrict__ contrib) {
    // xs transposed: smem[h*BT + r] ; reused after GEMM for reductions.
    __shared__ float smem[BT * HIDDEN];   // 64 KB

    const int e     = blockIdx.y;
    const int b0    = blockIdx.x * BT;
    const int tid   = threadIdx.x;
    const int wave  = tid >> 5;
    const int lane  = tid & 31;
    const int l15   = lane & 15;
    const int khalf = (lane >> 4) * 2;          // K row offset: 0 or 2
    const int rbase = (lane >> 4) * 8;          // C rows 0-7 vs 8-15

    // ---- stage x tile (transposed) into LDS, coalesced on global side ----
    for (int idx = tid; idx < BT * HIDDEN; idx += TPB) {
        int h = idx & (HIDDEN - 1);
        int r = idx >> 10;
        smem[h * BT + r] = x[(size_t)(b0 + r) * HIDDEN + h];
    }
    __syncthreads();

    const float* W1e  = W1 + (size_t)e * HIDDEN * DEXP;
    const int    dwav = wave * (TPW * 16);      // this wave's D base (256 cols)

    v8f acc[TPW];
#pragma unroll
    for (int t = 0; t < TPW; ++t)
        acc[t] = (v8f){0.f, 0.f, 0.f, 0.f, 0.f, 0.f, 0.f, 0.f};

    // ---- K loop: h = x[16 x 1024] (A), W1[e] slice (B), fp32 WMMA ----
    for (int h0 = 0; h0 < HIDDEN; h0 += 4) {
        const float* arow = &smem[(h0 + khalf) * BT + l15];
        v2f a;
        a.x = arow[0];       // A[M=l15][K=khalf]
        a.y = arow[BT];      // A[M=l15][K=khalf+1]
        const float* brow = W1e + (size_t)(h0 + khalf) * DEXP + dwav + l15;
#pragma unroll
        for (int t = 0; t < TPW; ++t) {
            v2f bb;
            bb.x = brow[t * 16];          // B[K=khalf  ][N]
            bb.y = brow[t * 16 + DEXP];   // B[K=khalf+1][N]
            acc[t] = __builtin_amdgcn_wmma_f32_16x16x4_f32(
                false, a, false, bb, (short)0, acc[t], false, false);
        }
    }

    // ---- layernorm statistics straight from accumulators ----
    const float* b1e = b1 + (size_t)e * DEXP;
    float b1v[TPW];
    float s1[8], s2[8];
#pragma unroll
    for (int v = 0; v < 8; ++v) { s1[v] = 0.f; s2[v] = 0.f; }
#pragma unroll
    for (int t = 0; t < TPW; ++t) {
        int d  = dwav + t * 16 + l15;
        float bv = b1e[d];
        b1v[t] = bv;
#pragma unroll
        for (int v = 0; v < 8; ++v) {
            float hv = acc[t][v] + bv;
            s1[v] += hv;
            s2[v] = fmaf(hv, hv, s2[v]);
        }
    }
    // reduce over the 16 lanes of each half (N dimension of this wave's slice)
#pragma unroll
    for (int v = 0; v < 8; ++v) {
#pragma unroll
        for (int m = 8; m >= 1; m >>= 1) {
            s1[v] += __shfl_xor(s1[v], m, 32);
            s2[v] += __shfl_xor(s2[v], m, 32);
        }
    }

    __syncthreads();                      // xs dead; reuse smem for reductions
    if (l15 == 0) {
#pragma unroll
        for (int v = 0; v < 8; ++v) {
            smem[wave * 16 + rbase + v]       = s1[v];
            smem[256 + wave * 16 + rbase + v] = s2[v];
        }
    }
    __syncthreads();
    if (tid < BT) {
        float ts1 = 0.f, ts2 = 0.f;
#pragma unroll
        for (int w = 0; w < WAVES; ++w) {
            ts1 += smem[w * 16 + tid];
            ts2 += smem[256 + w * 16 + tid];
        }
        float mu  = ts1 * (1.0f / DEXP);
        float var = ts2 * (1.0f / DEXP) - mu * mu;
        smem[512 + tid] = mu;
        smem[528 + tid] = rsqrtf(var + LN_EPS);
    }
    __syncthreads();

    // ---- normalize + relu + dot with W2 ----
    float mu[8], rs[8];
#pragma unroll
    for (int v = 0; v < 8; ++v) {
        mu[v] = smem[512 + rbase + v];
        rs[v] = smem[528 + rbase + v];
    }
    const float* lge = ln_g + (size_t)e * DEXP;
    const float* lbe = ln_b + (size_t)e * DEXP;
    const float* w2e = W2 + (size_t)e * DEXP;
    float dot[8];
#pragma unroll
    for (int v = 0; v < 8; ++v) dot[v] = 0.f;
#pragma unroll
    for (int t = 0; t < TPW; ++t) {
        int d = dwav + t * 16 + l15;
        float g   = lge[d];
        float lb  = lbe[d];
        float w2v = w2e[d];
#pragma unroll
        for (int v = 0; v < 8; ++v) {
            float hv = acc[t][v] + b1v[t];
            float hn = fmaf((hv - mu[v]) * rs[v], g, lb);
            hn = fmaxf(hn, 0.f);
            dot[v] = fmaf(hn, w2v, dot[v]);
        }
    }
#pragma unroll
    for (int v = 0; v < 8; ++v) {
#pragma unroll
        for (int m = 8; m >= 1; m >>= 1)
            dot[v] += __shfl_xor(dot[v], m, 32);
    }
    if (l15 == 0) {
#pragma unroll
        for (int v = 0; v < 8; ++v)
            smem[wave * 16 + rbase + v] = dot[v];
    }
    __syncthreads();
    if (tid < BT) {
        float s = 0.f;
#pragma unroll
        for (int w = 0; w < WAVES; ++w) s += smem[w * 16 + tid];
        float eo = tanhf(s + b2[e]);
        float gw = gate[(size_t)(b0 + tid) * NEXP + e];
        contrib[(size_t)(b0 + tid) * NEXP + e] = eo * gw;
    }
}

// ---------------------------------------------------------------------------
// Kernel 3: out[b] = sum_e contrib[b][e]
// ---------------------------------------------------------------------------
__global__ __launch_bounds__(256)
void moe_reduce_kernel(const float* __restrict__ contrib,
                       float* __restrict__ out) {
    int b = blockIdx.x * blockDim.x + threadIdx.x;
    if (b < BATCH) {
        const float* c = contrib + (size_t)b * NEXP;
        float s = 0.f;
#pragma unroll
        for (int e = 0; e < NEXP; ++e) s += c[e];
        out[b] = s;
    }
}

// ---------------------------------------------------------------------------
extern "C" void kernel_launch(void* const* d_in, const int* in_sizes, int n_in,
                              void* d_out, int out_size, void* d_ws, size_t ws_size,
                              hipStream_t stream) {
    (void)in_sizes; (void)n_in; (void)out_size; (void)ws_size;
    const float* x    = (const float*)d_in[0];  // [B, H]
    const float* Wg   = (const float*)d_in[1];  // [E, H]
    const float* bg   = (const float*)d_in[2];  // [E]
    const float* W1   = (const float*)d_in[3];  // [E, H, D]
    const float* b1   = (const float*)d_in[4];  // [E, D]
    const float* ln_g = (const float*)d_in[5];  // [E, D]
    const float* ln_b = (const float*)d_in[6];  // [E, D]
    const float* W2   = (const float*)d_in[7];  // [E, D]
    const float* b2   = (const float*)d_in[8];  // [E]
    float* out = (float*)d_out;                 // [B, 1]

    float* gate    = (float*)d_ws;              // [B, E]   (256 KB)
    float* contrib = gate + (size_t)BATCH * NEXP; // [B, E] (256 KB)

    // 1) gating softmax: 8 rows per 256-thread block
    moe_gating_kernel<<<BATCH / 8, 256, 0, stream>>>(x, Wg, bg, gate);

    // 2) fused experts: grid.x = batch tiles (fastest -> W1[e] L2 reuse),
    //    grid.y = expert
    dim3 grid(BATCH / BT, NEXP);
    moe_expert_kernel<<<grid, TPB, 0, stream>>>(x, W1, b1, ln_g, ln_b,
                                                W2, b2, gate, contrib);

    // 3) combine experts
    moe_reduce_kernel<<<(BATCH + 255) / 256, 256, 0, stream>>>(contrib, out);
}